// ContrastiveLoss_64278480551973
// MI455X (gfx1250) — compile-verified
//
#include <hip/hip_runtime.h>
#include <hip/hip_bf16.h>

// ---------------------------------------------------------------------------
// ContrastiveLoss (B=4096, D=256, T=0.07), fp32 in -> scalar fp32 out.
//   S[i,j] = (x_i . y_j)/T
//   loss   = sum_{i,k} log( cumsum_j<=k exp(S[i,j]) + (B-1-k) )
//          - sum_i (B-i) * S[i,i]
// f32 WMMA GEMM tiles + fused exp / DPP16 prefix-scan / log.
// y tiles double-buffered in LDS via GLOBAL_LOAD_ASYNC_TO_LDS_B128 (ASYNCcnt),
// overlapping tile t+1 copy with tile t compute. Single-wave workgroups.
// ---------------------------------------------------------------------------

typedef __attribute__((ext_vector_type(2))) float v2f;
typedef __attribute__((ext_vector_type(8))) float v8f;

#define B_N        4096
#define D_K        256
#define INV_T      14.2857142857142857f   // 1/0.07
#define NCHUNK     16                     // column chunks
#define JT_CHUNK   16                     // 16-col j-tiles per chunk (256 cols)
#define PITCH      260                    // LDS row pitch (dwords): 4*r banks, conflict-free
#define ROWS_WG    16                     // one wave <-> one 16-row tile

// --- CDNA5 async copy: 16B global -> LDS, tracked by ASYNCcnt --------------
__device__ __forceinline__ void async_ld16(const float* g, const float* l) {
    unsigned lds = (unsigned)(unsigned long long)l;   // generic shared ptr: [31:0] = LDS offset
    asm volatile("global_load_async_to_lds_b128 %0, %1, off"
                 :: "v"(lds), "v"(g) : "memory");
}
__device__ __forceinline__ void wait_async0() {
#if __has_builtin(__builtin_amdgcn_s_wait_asynccnt)
    __builtin_amdgcn_s_wait_asynccnt(0);
#else
    asm volatile("s_wait_asynccnt 0x0" ::: "memory");
#endif
}

// Inclusive prefix sum within each 16-lane row (DPP16 ROW_SHR, zero fill).
__device__ __forceinline__ float prefix16(float x) {
    float s = x;
    int b;
    b = __builtin_amdgcn_update_dpp(0, __float_as_int(s), 0x111, 0xf, 0xf, true); s += __int_as_float(b);
    b = __builtin_amdgcn_update_dpp(0, __float_as_int(s), 0x112, 0xf, 0xf, true); s += __int_as_float(b);
    b = __builtin_amdgcn_update_dpp(0, __float_as_int(s), 0x114, 0xf, 0xf, true); s += __int_as_float(b);
    b = __builtin_amdgcn_update_dpp(0, __float_as_int(s), 0x118, 0xf, 0xf, true); s += __int_as_float(b);
    return s;
}
// Broadcast lane 15/31 to its 16-lane group: src = (lane & 0x10) | 0x0F.
__device__ __forceinline__ float bcast15(float x) {
    return __int_as_float(__builtin_amdgcn_ds_swizzle(__float_as_int(x), 0x1F0));
}

// 16x16 S-tile GEMM over K=256: 64 x V_WMMA_F32_16X16X4_F32.
__device__ __forceinline__ v8f gemm_tile(const float* ax, const float* by) {
    v8f acc = {};
#pragma unroll
    for (int kc = 0; kc < 64; ++kc) {
        v2f a = *(const v2f*)(ax + kc * 4);
        v2f b = *(const v2f*)(by + kc * 4);
        acc = __builtin_amdgcn_wmma_f32_16x16x4_f32(
            false, a, false, b, (short)0, acc, false, false);
    }
    return acc;
}

__global__ void closs_zero(float* out) { if (threadIdx.x == 0) out[0] = 0.0f; }

// ---------------------------------------------------------------------------
// Pass 1: per-(row, chunk) sum of exp(S) over a 256-column chunk.
// grid = (256 row tiles, 16 chunks), block = 32 (one wave).
// ---------------------------------------------------------------------------
__global__ void closs_chunk_sums(const float* __restrict__ x,
                                 const float* __restrict__ y,
                                 float* __restrict__ ws) {
    __shared__ float xs[ROWS_WG * PITCH];
    __shared__ float ys[2 * 16 * PITCH];

    const int bg     = blockIdx.x;
    const int chunk  = blockIdx.y;
    const int lane   = threadIdx.x & 31;
    const int laneLo = lane & 15;
    const int hi     = lane >> 4;
    const int r0     = bg * ROWS_WG;

    // Async-stage x rows (16 x 256 f32) and first y tile.
    for (int q = lane; q < ROWS_WG * 64; q += 32) {
        int row = q >> 6, c4 = q & 63;
        async_ld16(x + (size_t)(r0 + row) * D_K + c4 * 4, &xs[row * PITCH + c4 * 4]);
    }
    {
        const int jbase = chunk * JT_CHUNK * 16;
        for (int q = lane; q < 16 * 64; q += 32) {
            int row = q >> 6, c4 = q & 63;
            async_ld16(y + (size_t)(jbase + row) * D_K + c4 * 4, &ys[row * PITCH + c4 * 4]);
        }
    }
    wait_async0();

    const float* ax = &xs[laneLo * PITCH + hi * 2];

    float sacc[8];
#pragma unroll
    for (int v = 0; v < 8; ++v) sacc[v] = 0.0f;

    for (int jt = 0; jt < JT_CHUNK; ++jt) {
        // Kick off async copy of tile jt+1 into the other buffer.
        if (jt + 1 < JT_CHUNK) {
            const int jn = (chunk * JT_CHUNK + jt + 1) * 16;
            float* dst = &ys[((jt + 1) & 1) * 16 * PITCH];
            for (int q = lane; q < 16 * 64; q += 32) {
                int row = q >> 6, c4 = q & 63;
                async_ld16(y + (size_t)(jn + row) * D_K + c4 * 4, &dst[row * PITCH + c4 * 4]);
            }
        }
        const float* by = &ys[(jt & 1) * 16 * PITCH + laneLo * PITCH + hi * 2];
        v8f acc = gemm_tile(ax, by);
#pragma unroll
        for (int v = 0; v < 8; ++v)
            sacc[v] += __expf(acc[v] * INV_T);
        wait_async0();                     // tile jt+1 landed; safe for next iter
    }

    // Reduce each 16-lane group (row sum); one lane per row writes ws.
#pragma unroll
    for (int v = 0; v < 8; ++v) {
        float r = sacc[v];
        r += __shfl_xor(r, 1);
        r += __shfl_xor(r, 2);
        r += __shfl_xor(r, 4);
        r += __shfl_xor(r, 8);
        if (laneLo == v)
            ws[(size_t)(r0 + v + hi * 8) * NCHUNK + chunk] = r;
    }
}

// ---------------------------------------------------------------------------
// Pass 2: per-row exclusive prefix over the 16 chunk sums.
// ---------------------------------------------------------------------------
__global__ void closs_chunk_prefix(float* __restrict__ ws) {
    int r = blockIdx.x * blockDim.x + threadIdx.x;
    if (r < B_N) {
        float run = 0.0f;
#pragma unroll
        for (int c = 0; c < NCHUNK; ++c) {
            float v = ws[(size_t)r * NCHUNK + c];
            ws[(size_t)r * NCHUNK + c] = run;
            run += v;
        }
    }
}

// ---------------------------------------------------------------------------
// Pass 3: recompute chunk tiles, exact scan with chunk carry, log, reduce.
// usePrefix==0: single sweep over all 4096 columns (fallback, chunk grid = 1).
// ---------------------------------------------------------------------------
__global__ void closs_main(const float* __restrict__ x,
                           const float* __restrict__ y,
                           const float* __restrict__ ws,
                           float* __restrict__ out,
                           int jtPerTask, int usePrefix) {
    __shared__ float xs[ROWS_WG * PITCH];
    __shared__ float ys[2 * 16 * PITCH];

    const int bg     = blockIdx.x;
    const int chunk  = blockIdx.y;
    const int lane   = threadIdx.x & 31;
    const int laneLo = lane & 15;
    const int hi     = lane >> 4;
    const int r0     = bg * ROWS_WG;

    for (int q = lane; q < ROWS_WG * 64; q += 32) {
        int row = q >> 6, c4 = q & 63;
        async_ld16(x + (size_t)(r0 + row) * D_K + c4 * 4, &xs[row * PITCH + c4 * 4]);
    }
    {
        const int jbase = chunk * jtPerTask * 16;
        for (int q = lane; q < 16 * 64; q += 32) {
            int row = q >> 6, c4 = q & 63;
            async_ld16(y + (size_t)(jbase + row) * D_K + c4 * 4, &ys[row * PITCH + c4 * 4]);
        }
    }

    // Per-row running carry (cumE of previous chunks); uniform per 16-lane group.
    float carry[8];
#pragma unroll
    for (int v = 0; v < 8; ++v) {
        carry[v] = 0.0f;
        if (usePrefix)
            carry[v] = ws[(size_t)(r0 + v + hi * 8) * NCHUNK + chunk];
    }
    wait_async0();

    const float* ax = &xs[laneLo * PITCH + hi * 2];
    float logSum = 0.0f, diagSum = 0.0f;

    for (int jt = 0; jt < jtPerTask; ++jt) {
        const int jbase = (chunk * jtPerTask + jt) * 16;
        if (jt + 1 < jtPerTask) {
            float* dst = &ys[((jt + 1) & 1) * 16 * PITCH];
            for (int q = lane; q < 16 * 64; q += 32) {
                int row = q >> 6, c4 = q & 63;
                async_ld16(y + (size_t)(jbase + 16 + row) * D_K + c4 * 4,
                           &dst[row * PITCH + c4 * 4]);
            }
        }
        const float* by = &ys[(jt & 1) * 16 * PITCH + laneLo * PITCH + hi * 2];
        v8f acc = gemm_tile(ax, by);

        const bool onDiag = (jbase == r0);                     // 16-aligned tiles
        const float cst = (float)(B_N - 1 - (jbase + laneLo)); // (B-1-k) term

#pragma unroll
        for (int v = 0; v < 8; ++v) {
            float s = acc[v] * INV_T;                          // S[i, j]
            if (onDiag) {
                int M = v + hi * 8;                            // row within tile
                if (laneLo == M)                               // diagonal element
                    diagSum += (float)(B_N - (r0 + M)) * s;    // n_hits * S[i,i]
            }
            float e  = __expf(s);
            float ps = prefix16(e);                            // in-tile inclusive cumsum
            float denom = carry[v] + ps + cst;
            logSum += __logf(denom);
            carry[v] += bcast15(ps);                           // tile total -> carry
        }
        wait_async0();
    }

    float part = logSum - diagSum;
    part += __shfl_xor(part, 1);
    part += __shfl_xor(part, 2);
    part += __shfl_xor(part, 4);
    part += __shfl_xor(part, 8);
    part += __shfl_xor(part, 16);
    if (lane == 0) atomicAdd(out, part);
}

// ---------------------------------------------------------------------------
extern "C" void kernel_launch(void* const* d_in, const int* in_sizes, int n_in,
                              void* d_out, int out_size, void* d_ws, size_t ws_size,
                              hipStream_t stream) {
    const float* x = (const float*)d_in[0];
    const float* y = (const float*)d_in[1];
    float* out = (float*)d_out;
    float* ws  = (float*)d_ws;

    closs_zero<<<1, 1, 0, stream>>>(out);

    const size_t wsNeeded = (size_t)B_N * NCHUNK * sizeof(float); // 256 KB
    const bool chunked = (ws_size >= wsNeeded);                   // fixed per run

    if (chunked) {
        closs_chunk_sums<<<dim3(B_N / ROWS_WG, NCHUNK), 32, 0, stream>>>(x, y, ws);
        closs_chunk_prefix<<<B_N / 256, 256, 0, stream>>>(ws);
        closs_main<<<dim3(B_N / ROWS_WG, NCHUNK), 32, 0, stream>>>(
            x, y, ws, out, JT_CHUNK, 1);
    } else {
        // Fallback: single sequential sweep over all 4096 columns per row tile.
        closs_main<<<dim3(B_N / ROWS_WG, 1), 32, 0, stream>>>(
            x, y, ws, out, B_N / 16, 0);
    }
}